// TwoLayerSimpleHeteroTAG_5265629905483
// MI455X (gfx1250) — compile-verified
//
#include <hip/hip_runtime.h>
#include <hip/hip_bf16.h>

// ---------------------------------------------------------------------------
// TwoLayerSimpleHeteroTAG for MI455X (gfx1250, wave32).
// Exploits: hop-2 of TAGConv is exactly zero on each bipartite relation, and
// src-side norms are 1.  So per relation/layer:
//   h1[dst]  = sum_{edges->dst} x[src]                (scatter-add SpMM)
//   out[dst] = x[dst] @ W[0:64] + (norm*h1)[dst] @ W[64:128] + b
// followed by per-type BatchNorm(batch stats) + SiLU.
// GEMM uses V_WMMA_F32_16X16X4_F32 (fp32 keeps reference precision; workload
// is HBM/atomic bound so fp32 WMMA is free).  K-loop is split into two
// branch-free phases so A-fragment loads pipeline cleanly.
// ---------------------------------------------------------------------------

#define NUU 200000
#define NII 200000
#define DD  64
#define EE  1000000

typedef __attribute__((ext_vector_type(2))) float v2f;
typedef __attribute__((ext_vector_type(8))) float v8f;

__device__ __forceinline__ void atomAddF(float* p, float v) {
#if defined(__gfx1250__)
    unsafeAtomicAdd(p, v);      // lowers to global_atomic_add_f32
#else
    atomicAdd(p, v);
#endif
}

// ---------------------------------------------------------------- zero fill
__global__ void zero_f32(float* __restrict__ p, long n) {
    long i = (long)blockIdx.x * blockDim.x + threadIdx.x;
    if (i < n) p[i] = 0.0f;
}

// ------------------------------------------------------------ degree count
__global__ void deg_count(const int* __restrict__ dst, float* __restrict__ deg, int nE) {
    int i = blockIdx.x * blockDim.x + threadIdx.x;
    if (i < nE) atomAddF(&deg[dst[i]], 1.0f);
}

// deg -> rsqrt(max(deg,1))
__global__ void norm_fin(float* __restrict__ d, int n) {
    int i = blockIdx.x * blockDim.x + threadIdx.x;
    if (i < n) d[i] = rsqrtf(fmaxf(d[i], 1.0f));
}

// -------------------------------------------------- SpMM scatter-add (hop 1)
// one thread handles (edge, 4 features): float4 gather + 4 f32 atomics
__global__ void scatter_add(const float* __restrict__ xsrc,
                            const int* __restrict__ src,
                            const int* __restrict__ dst,
                            float* __restrict__ h, long nwork) {
    long i = (long)blockIdx.x * blockDim.x + threadIdx.x;
    if (i >= nwork) return;
    long e  = i >> 4;
    int  c4 = (int)(i & 15) * 4;
    long s = src[e], d = dst[e];
    float4 v = *(const float4*)(xsrc + s * DD + c4);
    float* hp = h + d * DD + c4;
    atomAddF(hp + 0, v.x);
    atomAddF(hp + 1, v.y);
    atomAddF(hp + 2, v.z);
    atomAddF(hp + 3, v.w);
}

// ---------------------------------------------- fused TAGConv linear (WMMA)
// out[r,0:64] = xin[r,:] @ W[0:64,:] + (nrm[r]*h1[r,:]) @ W[64:128,:] + bias
// block = 128 threads = 4 waves; wave -> 16 rows; K=128 in steps of 4.
__launch_bounds__(128)
__global__ void tag_linear_wmma(const float* __restrict__ xin,
                                const float* __restrict__ h1,
                                const float* __restrict__ nrm,
                                const float* __restrict__ W,     // [>=128][64]
                                const float* __restrict__ bias,  // [64]
                                float* __restrict__ out,         // [NR][64]
                                int NR) {
    // W staged pair-interleaved: Wp[p*64+n] = (W[2p][n], W[2p+1][n])
    __shared__ float2 Wp[64 * 64];
    int t = threadIdx.x;
    for (int idx = t; idx < 64 * 64; idx += 128) {
        int p = idx >> 6, n = idx & 63;
        Wp[idx] = make_float2(W[(2 * p) * DD + n], W[(2 * p + 1) * DD + n]);
    }
    __syncthreads();

    int wave = t >> 5;
    int lane = t & 31;
    int half = lane >> 4;     // A/B: K sub-pair select (+2)
    int m    = lane & 15;     // A: row-in-tile / B,D: column-in-tile
    long rowbase = (long)blockIdx.x * 64 + wave * 16;

    const float* ax = xin + (rowbase + m) * DD + 2 * half;  // phase-1 A stream
    const float* ah = h1  + (rowbase + m) * DD + 2 * half;  // phase-2 A stream
    float sc0 = nrm[rowbase + m];
    v2f scale; scale.x = sc0; scale.y = sc0;

    v8f acc0 = {}, acc1 = {}, acc2 = {}, acc3 = {};

    // ---- phase 1: k in [0,64), A from xin (branch-free) ----
#pragma unroll
    for (int s = 0; s < 16; ++s) {
        v2f a = *(const v2f*)(ax + 4 * s);        // A[m][k0+2h .. +1]
        int p = 2 * s + half;                     // pair row in Wp
        float2 b0 = Wp[p * 64 +  0 + m];
        float2 b1 = Wp[p * 64 + 16 + m];
        float2 b2 = Wp[p * 64 + 32 + m];
        float2 b3 = Wp[p * 64 + 48 + m];
        v2f vb0; vb0.x = b0.x; vb0.y = b0.y;
        v2f vb1; vb1.x = b1.x; vb1.y = b1.y;
        v2f vb2; vb2.x = b2.x; vb2.y = b2.y;
        v2f vb3; vb3.x = b3.x; vb3.y = b3.y;
        acc0 = __builtin_amdgcn_wmma_f32_16x16x4_f32(false, a, false, vb0, (short)0, acc0, false, false);
        acc1 = __builtin_amdgcn_wmma_f32_16x16x4_f32(false, a, false, vb1, (short)0, acc1, false, false);
        acc2 = __builtin_amdgcn_wmma_f32_16x16x4_f32(false, a, false, vb2, (short)0, acc2, false, false);
        acc3 = __builtin_amdgcn_wmma_f32_16x16x4_f32(false, a, false, vb3, (short)0, acc3, false, false);
    }

    // ---- phase 2: k in [64,128), A = norm * h1 (branch-free) ----
#pragma unroll
    for (int s = 0; s < 16; ++s) {
        v2f a = *(const v2f*)(ah + 4 * s);
        a = a * scale;                            // v_pk_mul_f32
        int p = 32 + 2 * s + half;
        float2 b0 = Wp[p * 64 +  0 + m];
        float2 b1 = Wp[p * 64 + 16 + m];
        float2 b2 = Wp[p * 64 + 32 + m];
        float2 b3 = Wp[p * 64 + 48 + m];
        v2f vb0; vb0.x = b0.x; vb0.y = b0.y;
        v2f vb1; vb1.x = b1.x; vb1.y = b1.y;
        v2f vb2; vb2.x = b2.x; vb2.y = b2.y;
        v2f vb3; vb3.x = b3.x; vb3.y = b3.y;
        acc0 = __builtin_amdgcn_wmma_f32_16x16x4_f32(false, a, false, vb0, (short)0, acc0, false, false);
        acc1 = __builtin_amdgcn_wmma_f32_16x16x4_f32(false, a, false, vb1, (short)0, acc1, false, false);
        acc2 = __builtin_amdgcn_wmma_f32_16x16x4_f32(false, a, false, vb2, (short)0, acc2, false, false);
        acc3 = __builtin_amdgcn_wmma_f32_16x16x4_f32(false, a, false, vb3, (short)0, acc3, false, false);
    }

    float bz0 = bias[ 0 + m];
    float bz1 = bias[16 + m];
    float bz2 = bias[32 + m];
    float bz3 = bias[48 + m];

    // D layout: VGPR j -> M = j (lanes 0-15) / M = 8+j (lanes 16-31); N = lane&15
#pragma unroll
    for (int j = 0; j < 8; ++j) {
        long r = rowbase + half * 8 + j;
        float* o = out + r * DD + m;
        o[ 0] = acc0[j] + bz0;
        o[16] = acc1[j] + bz1;
        o[32] = acc2[j] + bz2;
        o[48] = acc3[j] + bz3;
    }
}

// ------------------------------------------------------------ BN batch stats
// block = 256 = 64 cols x 4 row-threads; partial sums -> LDS -> global atomics
__global__ void bn_stats(const float* __restrict__ x,
                         float* __restrict__ sum, float* __restrict__ sq, int NR) {
    __shared__ float ssum[256], ssq[256];
    int c  = threadIdx.x & 63;
    int rs = threadIdx.x >> 6;
    float s = 0.0f, q = 0.0f;
    for (long r = (long)blockIdx.x * 4 + rs; r < NR; r += (long)gridDim.x * 4) {
        float v = x[r * DD + c];
        s += v;
        q += v * v;
    }
    ssum[threadIdx.x] = s;
    ssq[threadIdx.x]  = q;
    __syncthreads();
    if (rs == 0) {
        s = ssum[c] + ssum[c + 64] + ssum[c + 128] + ssum[c + 192];
        q = ssq[c]  + ssq[c + 64]  + ssq[c + 128]  + ssq[c + 192];
        atomAddF(&sum[c], s);
        atomAddF(&sq[c],  q);
    }
}

// --------------------------------------------------- BN apply + SiLU (inplace)
__global__ void bn_apply_silu(float* __restrict__ x,
                              const float* __restrict__ sum, const float* __restrict__ sq,
                              const float* __restrict__ g, const float* __restrict__ be,
                              int NR) {
    long i = (long)blockIdx.x * blockDim.x + threadIdx.x;
    long n = (long)NR * DD;
    if (i >= n) return;
    int c = (int)(i & 63);
    float inv = 1.0f / (float)NR;
    float mu  = sum[c] * inv;
    float var = sq[c] * inv - mu * mu;
    float y = (x[i] - mu) * rsqrtf(var + 1e-5f) * g[c] + be[c];
    x[i] = y / (1.0f + __expf(-y));
}

// ---------------------------------------------------------------------------
extern "C" void kernel_launch(void* const* d_in, const int* in_sizes, int n_in,
                              void* d_out, int out_size, void* d_ws, size_t ws_size,
                              hipStream_t stream) {
    const float* user_emb = (const float*)d_in[0];
    const float* item_emb = (const float*)d_in[1];
    const float* W_go1    = (const float*)d_in[2];
    const float* b_go1    = (const float*)d_in[3];
    const float* W_back1  = (const float*)d_in[4];
    const float* b_back1  = (const float*)d_in[5];
    const float* W_go2    = (const float*)d_in[6];
    const float* b_go2    = (const float*)d_in[7];
    const float* W_back2  = (const float*)d_in[8];
    const float* b_back2  = (const float*)d_in[9];
    const float* g1u  = (const float*)d_in[10];
    const float* be1u = (const float*)d_in[11];
    const float* g1i  = (const float*)d_in[12];
    const float* be1i = (const float*)d_in[13];
    const float* g2u  = (const float*)d_in[14];
    const float* be2u = (const float*)d_in[15];
    const float* g2i  = (const float*)d_in[16];
    const float* be2i = (const float*)d_in[17];
    const int* go_src   = (const int*)d_in[18];
    const int* go_dst   = (const int*)d_in[19];
    const int* back_src = (const int*)d_in[20];
    const int* back_dst = (const int*)d_in[21];
    float* out = (float*)d_out;

    // workspace layout (floats)
    float* ws        = (float*)d_ws;
    float* norm_go   = ws;                           // NII  (item in-deg under go)
    float* norm_back = norm_go + NII;                // NUU  (user in-deg under back)
    float* h1_item   = norm_back + NUU;              // NII*64
    float* h1_user   = h1_item + (long)NII * DD;     // NUU*64
    float* bn        = h1_user + (long)NUU * DD;     // 8 x 64 accumulators
    float* x2        = bn + 512;                     // N*64: user part then item part
    float* x2_user   = x2;
    float* x2_item   = x2 + (long)NUU * DD;
    float* out_user  = out;
    float* out_item  = out + (long)NUU * DD;

    const int B = 256;
    long zc1 = (long)(NII + NUU) + (long)(NII + NUU) * DD + 512;   // norms+h1+bn
    long h1c = (long)(NII + NUU) * DD;
    long sc  = (long)EE * 16;
    long apc = (long)NUU * DD;   // == NII*DD

    // ---- structural prep ----
    zero_f32<<<dim3((unsigned)((zc1 + B - 1) / B)), B, 0, stream>>>(ws, zc1);
    deg_count<<<dim3((EE + B - 1) / B), B, 0, stream>>>(go_dst, norm_go, EE);
    deg_count<<<dim3((EE + B - 1) / B), B, 0, stream>>>(back_dst, norm_back, EE);
    norm_fin<<<dim3((NII + NUU + B - 1) / B), B, 0, stream>>>(norm_go, NII + NUU);

    // ---- layer 1 ----
    scatter_add<<<dim3((unsigned)((sc + B - 1) / B)), B, 0, stream>>>(user_emb, go_src, go_dst, h1_item, sc);
    scatter_add<<<dim3((unsigned)((sc + B - 1) / B)), B, 0, stream>>>(item_emb, back_src, back_dst, h1_user, sc);
    tag_linear_wmma<<<dim3(NII / 64), 128, 0, stream>>>(item_emb, h1_item, norm_go,   W_go1,   b_go1,   x2_item, NII);
    tag_linear_wmma<<<dim3(NUU / 64), 128, 0, stream>>>(user_emb, h1_user, norm_back, W_back1, b_back1, x2_user, NUU);
    bn_stats<<<dim3(512), 256, 0, stream>>>(x2_user, bn + 0,   bn + 64,  NUU);
    bn_apply_silu<<<dim3((unsigned)((apc + B - 1) / B)), B, 0, stream>>>(x2_user, bn + 0,   bn + 64,  g1u, be1u, NUU);
    bn_stats<<<dim3(512), 256, 0, stream>>>(x2_item, bn + 128, bn + 192, NII);
    bn_apply_silu<<<dim3((unsigned)((apc + B - 1) / B)), B, 0, stream>>>(x2_item, bn + 128, bn + 192, g1i, be1i, NII);

    // ---- layer 2 ----
    zero_f32<<<dim3((unsigned)((h1c + B - 1) / B)), B, 0, stream>>>(h1_item, h1c);
    scatter_add<<<dim3((unsigned)((sc + B - 1) / B)), B, 0, stream>>>(x2_user, go_src, go_dst, h1_item, sc);
    scatter_add<<<dim3((unsigned)((sc + B - 1) / B)), B, 0, stream>>>(x2_item, back_src, back_dst, h1_user, sc);
    tag_linear_wmma<<<dim3(NII / 64), 128, 0, stream>>>(x2_item, h1_item, norm_go,   W_go2,   b_go2,   out_item, NII);
    tag_linear_wmma<<<dim3(NUU / 64), 128, 0, stream>>>(x2_user, h1_user, norm_back, W_back2, b_back2, out_user, NUU);
    bn_stats<<<dim3(512), 256, 0, stream>>>(out_user, bn + 256, bn + 320, NUU);
    bn_apply_silu<<<dim3((unsigned)((apc + B - 1) / B)), B, 0, stream>>>(out_user, bn + 256, bn + 320, g2u, be2u, NUU);
    bn_stats<<<dim3(512), 256, 0, stream>>>(out_item, bn + 384, bn + 448, NII);
    bn_apply_silu<<<dim3((unsigned)((apc + B - 1) / B)), B, 0, stream>>>(out_item, bn + 384, bn + 448, g2i, be2i, NII);
}